// MultiHeadAttention_57217554317695
// MI455X (gfx1250) — compile-verified
//
#include <hip/hip_runtime.h>
#include <hip/hip_bf16.h>

typedef __attribute__((ext_vector_type(2))) float v2f;
typedef __attribute__((ext_vector_type(8))) float v8f;

static constexpr int Bsz = 2, Tsz = 2048, Dsz = 1024, Hn = 16, Kd = 64;
static constexpr int MROWS = Bsz * Tsz;   // 4096

#if defined(__has_builtin)
#if __has_builtin(__builtin_amdgcn_global_load_async_to_lds_b128)
#define HAVE_ASYNC_LDS 1
#endif
#if __has_builtin(__builtin_amdgcn_tensor_load_to_lds)
#define HAVE_TDM 1
#endif
#endif

typedef unsigned int u32x4 __attribute__((vector_size(16)));
typedef int i32x8 __attribute__((vector_size(32)));
typedef int i32x4 __attribute__((vector_size(16)));

// 16-byte global -> LDS copy: async (GLOBAL_LOAD_ASYNC_TO_LDS_B128) if the
// builtin exists, else synchronous bounce through VGPRs.
__device__ __forceinline__ void copy16_g2l(float* lds, const float* g) {
#ifdef HAVE_ASYNC_LDS
  typedef int vint4 __attribute__((vector_size(16)));
  typedef __attribute__((address_space(1))) vint4 as1_vint4;
  typedef __attribute__((address_space(3))) vint4 as3_vint4;
  __builtin_amdgcn_global_load_async_to_lds_b128(
      (as1_vint4*)g, (as3_vint4*)lds, 0, 0);
#else
  *(float4*)lds = *(const float4*)g;
#endif
}

__device__ __forceinline__ void wait_async0() {
#ifdef HAVE_ASYNC_LDS
#if __has_builtin(__builtin_amdgcn_s_wait_asynccnt)
  __builtin_amdgcn_s_wait_asynccnt(0);
#else
  asm volatile("s_wait_asynccnt 0" ::: "memory");
#endif
#endif
}

__device__ __forceinline__ void wait_tensor0() {
#if __has_builtin(__builtin_amdgcn_s_wait_tensorcnt)
  __builtin_amdgcn_s_wait_tensorcnt(0);
#else
  asm volatile("s_wait_tensorcnt 0" ::: "memory");
#endif
}

__device__ __forceinline__ unsigned lds_byte_addr(const void* p) {
  typedef __attribute__((address_space(3))) const void as3_void;
  return (unsigned)(uintptr_t)(as3_void*)p;
}

#ifdef HAVE_TDM
// One TDM descriptor: load a 2-D tile (tile_d1 rows x tile_d0 dwords) from a
// row-major f32 tensor (row stride d0_stride dwords) into LDS at lds_off,
// inserting pad_amt dwords of LDS padding every (1<<(pad_iv+1)) dwords.
// D# layout per CDNA5 ISA.
__device__ __forceinline__ void tdm_load_2d(const void* gsrc, unsigned lds_off,
                                            unsigned tile_d0, unsigned tile_d1,
                                            unsigned tensor_d0, unsigned tensor_d1,
                                            unsigned long long d0_stride,
                                            unsigned pad_iv, unsigned pad_amt_m1) {
  unsigned long long ga = (unsigned long long)(uintptr_t)gsrc;
  u32x4 g0;
  g0[0] = 1u;                                   // count=1, user mode, no gather
  g0[1] = lds_off;                              // lds_addr (bytes)
  g0[2] = (unsigned)(ga & 0xFFFFFFFFu);         // global_addr[31:0]
  g0[3] = (unsigned)((ga >> 32) & 0x1FFFFFFu)   // global_addr[56:32]
          | (2u << 30);                         // type = 2 ("image")
  i32x8 g1;
  unsigned w0 = (2u << 16)                      // data_size = 4 bytes
              | (1u << 20)                      // pad_enable
              | ((pad_iv & 7u) << 22)           // pad_interval: 2^(v+1) dwords
              | ((pad_amt_m1 & 127u) << 25);    // pad_amount: (a+1) dwords
  g1[0] = (int)w0;                              // workgroup_mask=0 (no cluster)
  g1[1] = (int)((tensor_d0 & 0xFFFFu) << 16);   // [15:0] atomic_barrier=0, [31:16] tdim0.lo
  g1[2] = (int)((tensor_d0 >> 16) | ((tensor_d1 & 0xFFFFu) << 16));
  g1[3] = (int)((tensor_d1 >> 16) | (tile_d0 << 16));
  g1[4] = (int)(tile_d1 & 0xFFFFu);             // tile_dim1; tile_dim2=0 (2D)
  g1[5] = (int)(d0_stride & 0xFFFFFFFFull);     // tensor_dim0_stride[31:0]
  g1[6] = (int)((d0_stride >> 32) & 0xFFFFull); // stride[47:32]; dim1_stride=0
  g1[7] = 0;
  i32x4 z4 = {0, 0, 0, 0};                      // groups 2/3 unused (<=2D tile)
  i32x8 z8 = {0, 0, 0, 0, 0, 0, 0, 0};
  __builtin_amdgcn_tensor_load_to_lds(g0, g1, z4, z4, z8, 0);
}
#endif

__device__ __forceinline__ v8f wmma_f32(v2f a, v2f b, v8f c) {
  // D = A(16x4,f32) x B(4x16,f32) + C(16x16,f32)
  return __builtin_amdgcn_wmma_f32_16x16x4_f32(
      /*neg_a=*/false, a, /*neg_b=*/false, b,
      /*c_mod=*/(short)0, c, /*reuse_a=*/false, /*reuse_b=*/false);
}

// ---------------------------------------------------------------------------
// Generic projection GEMM: Y = X[4096x1024] * W[1024x1024] + bias[1024]
// remap==1: scatter to [B,H,T,64] (for Q/K/V).  remap==0: row-major.
// Block = 256 threads (8 waves), block tile 128x64, K-tile 32, double-buffered
// TDM (or async-LDS) staging.
// ---------------------------------------------------------------------------
__global__ __launch_bounds__(256) void proj_gemm(
    const float* __restrict__ X, const float* __restrict__ W,
    const float* __restrict__ bias, float* __restrict__ Y, int remap) {
  __shared__ float As[2][128][36];  // stride 36 = 32 + pad 4: conflict-free A-frags
  __shared__ float Bs[2][32][72];   // stride 72 = 64 + pad 8: conflict-free B-frags

  const int tid  = threadIdx.x;
  const int wid  = tid >> 5;
  const int lane = tid & 31;
  const int half = lane >> 4;
  const int l16  = lane & 15;
  const int m0   = blockIdx.y * 128;
  const int n0   = blockIdx.x * 64;

  v8f acc[4] = {};

  auto stage = [&](int kt0, int buf) {
#ifdef HAVE_TDM
    if (wid == 0) {  // one TDM descriptor per tile, issued by wave 0 (uniform)
      tdm_load_2d(&X[(size_t)m0 * Dsz + kt0], lds_byte_addr(&As[buf][0][0]),
                  /*tile*/ 32, 128, /*tensor*/ Dsz, MROWS, /*stride*/ Dsz,
                  /*pad: +4 dw every 32 dw*/ 4, 3);
      tdm_load_2d(&W[(size_t)kt0 * Dsz + n0], lds_byte_addr(&Bs[buf][0][0]),
                  /*tile*/ 64, 32, /*tensor*/ Dsz, Dsz, /*stride*/ Dsz,
                  /*pad: +8 dw every 64 dw*/ 5, 7);
    }
#else
#pragma unroll
    for (int j = 0; j < 4; ++j) {  // A tile 128x32: 4 x b128 per thread
      int idx = tid + j * 256;
      int r = idx >> 3, c = (idx & 7) * 4;
      copy16_g2l(&As[buf][r][c], &X[(size_t)(m0 + r) * Dsz + kt0 + c]);
    }
#pragma unroll
    for (int j = 0; j < 2; ++j) {  // B tile 32x64: 2 x b128 per thread
      int idx = tid + j * 256;
      int r = idx >> 4, c = (idx & 15) * 4;
      copy16_g2l(&Bs[buf][r][c], &W[(size_t)(kt0 + r) * Dsz + n0 + c]);
    }
#endif
  };
  auto stage_wait = [&]() {
#ifdef HAVE_TDM
    if (wid == 0) wait_tensor0();
#else
    wait_async0();
#endif
    __syncthreads();
  };

  int buf = 0;
  stage(0, 0);
  stage_wait();

  for (int kt0 = 0; kt0 < Dsz; kt0 += 32) {
    if (kt0 + 32 < Dsz) stage(kt0 + 32, buf ^ 1);  // async prefetch next tile

    const float(*as)[36] = As[buf];
    const float(*bs)[72] = Bs[buf];
#pragma unroll
    for (int kk = 0; kk < 32; kk += 4) {
      // A-frag (16x4): lanes 0-15 hold K=kk,kk+1; lanes 16-31 hold K=kk+2,kk+3
      v2f a = *(const v2f*)&as[wid * 16 + l16][kk + 2 * half];
#pragma unroll
      for (int t = 0; t < 4; ++t) {
        v2f b;
        b.x = bs[kk + 2 * half][t * 16 + l16];
        b.y = bs[kk + 1 + 2 * half][t * 16 + l16];
        acc[t] = wmma_f32(a, b, acc[t]);
      }
    }

    stage_wait();
    buf ^= 1;
  }

  // epilogue: bias + store (C layout: vgpr j -> row j+8*half, col = l16)
#pragma unroll
  for (int t = 0; t < 4; ++t) {
#pragma unroll
    for (int j = 0; j < 8; ++j) {
      int m = m0 + wid * 16 + j + 8 * half;
      int n = n0 + t * 16 + l16;
      float val = acc[t][j] + bias[n];
      if (remap) {  // [B,T,H*K] -> [B,H,T,Kd]
        int bb = m >> 11, tt = m & (Tsz - 1);
        int hh = n >> 6, kk = n & (Kd - 1);
        Y[((size_t)(bb * Hn + hh) * Tsz + tt) * Kd + kk] = val;
      } else {
        Y[(size_t)m * Dsz + n] = val;
      }
    }
  }
}

// ---------------------------------------------------------------------------
// Flash attention: Q,K,V in [B,H,T,64]; output [B,T,H*64].
// One wave = one 16-query tile; 8 waves/block share (b,h); K/V 16x64 tiles
// staged via double-buffered TDM (or async) copies.  Online softmax, f32 WMMA.
// ---------------------------------------------------------------------------
__global__ __launch_bounds__(256) void flash_attn(
    const float* __restrict__ Q, const float* __restrict__ Kt,
    const float* __restrict__ V, float* __restrict__ O) {
  __shared__ float Ks[2][16][68];  // stride 68 = 64 + pad 4
  __shared__ float Vs[2][16][68];
  __shared__ float Ps[8][16][18];  // per-wave 16x16 P transpose buffer

  const int tid  = threadIdx.x;
  const int wid  = tid >> 5;
  const int lane = tid & 31;
  const int half = lane >> 4;
  const int l16  = lane & 15;

  const int gw = blockIdx.x * 8 + wid;  // global wave id
  const int bh = gw >> 7;               // 128 q-tiles per (b,h); uniform in block
  const int qt = gw & 127;

  const float* qbase = Q  + (size_t)bh * Tsz * Kd;
  const float* kbase = Kt + (size_t)bh * Tsz * Kd;
  const float* vbase = V  + (size_t)bh * Tsz * Kd;

  // Q fragments held in registers for all 16 K-steps (cols 4i+2*half, +1)
  v2f qf[16];
  {
    const float* qrow = qbase + (size_t)(qt * 16 + l16) * Kd + 2 * half;
#pragma unroll
    for (int i = 0; i < 16; ++i) qf[i] = *(const v2f*)&qrow[4 * i];
  }

  v8f o[4] = {};
  float mrow[8], lrow[8];
#pragma unroll
  for (int j = 0; j < 8; ++j) { mrow[j] = -__builtin_inff(); lrow[j] = 0.f; }

  const int sr = tid >> 4, sc = (tid & 15) * 4;  // coop staging coords
  auto stage = [&](int kt, int buf) {
#ifdef HAVE_TDM
    if (wid == 0) {
      tdm_load_2d(&kbase[(size_t)kt * 16 * Kd], lds_byte_addr(&Ks[buf][0][0]),
                  /*tile*/ Kd, 16, /*tensor*/ Kd, Tsz, /*stride*/ Kd,
                  /*pad: +4 dw every 64 dw*/ 5, 3);
      tdm_load_2d(&vbase[(size_t)kt * 16 * Kd], lds_byte_addr(&Vs[buf][0][0]),
                  Kd, 16, Kd, Tsz, Kd, 5, 3);
    }
#else
    copy16_g2l(&Ks[buf][sr][sc], &kbase[(size_t)(kt * 16 + sr) * Kd + sc]);
    copy16_g2l(&Vs[buf][sr][sc], &vbase[(size_t)(kt * 16 + sr) * Kd + sc]);
#endif
  };
  auto stage_wait = [&]() {
#ifdef HAVE_TDM
    if (wid == 0) wait_tensor0();
#else
    wait_async0();
#endif
    __syncthreads();
  };

  int buf = 0;
  stage(0, 0);
  stage_wait();

  constexpr int NT = Tsz / 16;
  for (int kt = 0; kt < NT; ++kt) {
    if (kt + 1 < NT) stage(kt + 1, buf ^ 1);  // async prefetch next K/V tile

    const float(*ks)[68] = Ks[buf];
    const float(*vs)[68] = Vs[buf];

    // S = Q * K^T  (16x16 tile; B[k][n] = Ks[n][k])
    v8f s = {};
#pragma unroll
    for (int i = 0; i < 16; ++i) {
      v2f bk = *(const v2f*)&ks[l16][4 * i + 2 * half];
      s = wmma_f32(qf[i], bk, s);
    }

    // online softmax per row (row of vgpr j is striped across 16 lanes)
#pragma unroll
    for (int j = 0; j < 8; ++j) {
      float v = s[j] * 0.125f;  // 1/sqrt(64)
      float vmax = v;
      vmax = fmaxf(vmax, __shfl_xor(vmax, 1, 16));
      vmax = fmaxf(vmax, __shfl_xor(vmax, 2, 16));
      vmax = fmaxf(vmax, __shfl_xor(vmax, 4, 16));
      vmax = fmaxf(vmax, __shfl_xor(vmax, 8, 16));
      float mnew  = fmaxf(mrow[j], vmax);
      float alpha = __expf(mrow[j] - mnew);
      float p     = __expf(v - mnew);
      float psum = p;
      psum += __shfl_xor(psum, 1, 16);
      psum += __shfl_xor(psum, 2, 16);
      psum += __shfl_xor(psum, 4, 16);
      psum += __shfl_xor(psum, 8, 16);
      lrow[j] = lrow[j] * alpha + psum;
      mrow[j] = mnew;
      s[j] = p;
#pragma unroll
      for (int t = 0; t < 4; ++t) o[t][j] *= alpha;
    }

    // C-layout -> A-layout for P via per-wave LDS bounce (wave-local ordering)
#pragma unroll
    for (int j = 0; j < 8; ++j) Ps[wid][j + 8 * half][l16] = s[j];
    asm volatile("s_wait_dscnt 0" ::: "memory");

    // O += P * V
#pragma unroll
    for (int i = 0; i < 4; ++i) {
      v2f ap = *(const v2f*)&Ps[wid][l16][4 * i + 2 * half];
#pragma unroll
      for (int t = 0; t < 4; ++t) {
        v2f bv;
        bv.x = vs[4 * i + 2 * half][t * 16 + l16];
        bv.y = vs[4 * i + 1 + 2 * half][t * 16 + l16];
        o[t] = wmma_f32(ap, bv, o[t]);
      }
    }

    stage_wait();
    buf ^= 1;
  }

  // normalize and store into [B, T, H*Kd]
  const int bb = bh >> 4, hh = bh & 15;
#pragma unroll
  for (int j = 0; j < 8; ++j) {
    float inv = 1.f / lrow[j];
    int m = qt * 16 + j + 8 * half;
    size_t rowoff = ((size_t)(bb * Tsz + m)) * Dsz + hh * Kd;
#pragma unroll
    for (int t = 0; t < 4; ++t) O[rowoff + t * 16 + l16] = o[t][j] * inv;
  }
}

// ---------------------------------------------------------------------------
extern "C" void kernel_launch(void* const* d_in, const int* in_sizes, int n_in,
                              void* d_out, int out_size, void* d_ws, size_t ws_size,
                              hipStream_t stream) {
  const float* x  = (const float*)d_in[0];
  const float* Wq = (const float*)d_in[1];
  const float* bq = (const float*)d_in[2];
  const float* Wk = (const float*)d_in[3];
  const float* bk = (const float*)d_in[4];
  const float* Wv = (const float*)d_in[5];
  const float* bv = (const float*)d_in[6];
  const float* Wo = (const float*)d_in[7];
  const float* bo = (const float*)d_in[8];
  float* out = (float*)d_out;

  float* ws = (float*)d_ws;
  const size_t nmat = (size_t)MROWS * Dsz;  // 4096*1024
  float* qws = ws;
  float* kws = ws + nmat;
  float* vws = ws + 2 * nmat;
  float* aws = ws + 3 * nmat;

  dim3 gGemm(Dsz / 64, MROWS / 128);  // 16 x 32
  dim3 blk(256);

  proj_gemm<<<gGemm, blk, 0, stream>>>(x, Wq, bq, qws, 1);
  proj_gemm<<<gGemm, blk, 0, stream>>>(x, Wk, bk, kws, 1);
  proj_gemm<<<gGemm, blk, 0, stream>>>(x, Wv, bv, vws, 1);

  flash_attn<<<dim3((Bsz * Hn * (Tsz / 16)) / 8), blk, 0, stream>>>(qws, kws, vws, aws);

  proj_gemm<<<gGemm, blk, 0, stream>>>(aws, Wo, bo, out, 0);
}